// MaskedAutoencoderViT_83219286327800
// MI455X (gfx1250) — compile-verified
//
#include <hip/hip_runtime.h>
#include <math.h>
#include <stdint.h>

typedef __attribute__((ext_vector_type(2))) float v2f;
typedef __attribute__((ext_vector_type(8))) float v8f;

#define BATCH   32
#define SEQ     2000
#define DMODEL  1024
#define NKEEP   500
#define NMASK   (SEQ - NKEEP)               // 1500
#define NROWS   (BATCH * (NKEEP + 1))       // 16032 rows of latent / xd
#define LOSS_NB 128                          // partial-reduction blocks

// ---------------------------------------------------------------------------
// CDNA5 async global->LDS copies (per-lane), tracked with ASYNCcnt
// ---------------------------------------------------------------------------
__device__ __forceinline__ void async_copy_b128(uint32_t lds_addr,
                                                const float* gptr) {
  asm volatile("global_load_async_to_lds_b128 %0, %1, off"
               :
               : "v"(lds_addr), "v"(gptr)
               : "memory");
}

__device__ __forceinline__ void async_copy_b32(uint32_t lds_addr,
                                               const float* gptr) {
  asm volatile("global_load_async_to_lds_b32 %0, %1, off"
               :
               : "v"(lds_addr), "v"(gptr)
               : "memory");
}

__device__ __forceinline__ void wait_asynccnt0() {
  asm volatile("s_wait_asynccnt 0x0" ::: "memory");
}

// low 32 bits of a flat shared-memory address == wave-relative LDS offset
__device__ __forceinline__ uint32_t lds_off(const void* p) {
  return (uint32_t)(uintptr_t)p;
}

// ---------------------------------------------------------------------------
// block-wide sum over 256 threads (LDS tree; deterministic)
// ---------------------------------------------------------------------------
__device__ __forceinline__ float block_sum_256(float v, float* sbuf) {
  const int tid = threadIdx.x;
  __syncthreads();                 // protect previous use of sbuf
  sbuf[tid] = v;
  __syncthreads();
  for (int s = 128; s > 0; s >>= 1) {
    if (tid < s) sbuf[tid] += sbuf[tid + s];
    __syncthreads();
  }
  return sbuf[0];
}

// ---------------------------------------------------------------------------
// 1) stable rank per batch row -> shuffle, restore(=rank), mask
//    jnp.argsort is stable: rank = #{j : n[j]<n[i] or (n[j]==n[i] and j<i)}
// ---------------------------------------------------------------------------
__global__ __launch_bounds__(256) void rank_kernel(
    const float* __restrict__ noise, int* __restrict__ shuffle,
    int* __restrict__ restore, float* __restrict__ mask_out) {
  __shared__ float sn[SEQ];
  const int b = blockIdx.x, tid = threadIdx.x;
  for (int i = tid; i < SEQ; i += 256) sn[i] = noise[b * SEQ + i];
  __syncthreads();
  for (int i = tid; i < SEQ; i += 256) {
    const float vi = sn[i];
    int r = 0;
    for (int j = 0; j < SEQ; ++j) {
      const float vj = sn[j];
      r += (vj < vi) || (vj == vi && j < i);
    }
    restore[b * SEQ + i]  = r;
    shuffle[b * SEQ + r]  = i;
    mask_out[b * SEQ + i] = (r < NKEEP) ? 0.0f : 1.0f;
  }
}

// ---------------------------------------------------------------------------
// 2) build h row (cls or gathered token) + LayerNorm(gamma1,beta1) -> latent
//    one block per row; also emit latent[:,0] output
// ---------------------------------------------------------------------------
__global__ __launch_bounds__(256) void build_latent_kernel(
    const float* __restrict__ expr, const int* __restrict__ idx,
    const float* __restrict__ pos, const float* __restrict__ cls,
    const float* __restrict__ w_enc, const float* __restrict__ b_enc,
    const float* __restrict__ g1, const float* __restrict__ be1,
    const int* __restrict__ shuffle, float* __restrict__ latent,
    float* __restrict__ lat0_out) {
  __shared__ float sbuf[256];
  const int blk = blockIdx.x;               // 0..NROWS-1
  const int b = blk / (NKEEP + 1);
  const int n = blk % (NKEEP + 1);
  const int tid = threadIdx.x;

  float vals[4];
  if (n == 0) {
    #pragma unroll
    for (int i = 0; i < 4; ++i) {
      const int d = tid + 256 * i;
      vals[i] = cls[d] + pos[d];            // cls_token + pos_table[0]
    }
  } else {
    const int k = shuffle[b * SEQ + (n - 1)];
    const float e = expr[b * SEQ + k];
    const int p = idx[b * SEQ + k];
    #pragma unroll
    for (int i = 0; i < 4; ++i) {
      const int d = tid + 256 * i;
      vals[i] = e * w_enc[d] + b_enc[d] + pos[(size_t)p * DMODEL + d];
    }
  }
  float s = vals[0] + vals[1] + vals[2] + vals[3];
  const float mean = block_sum_256(s, sbuf) * (1.0f / DMODEL);
  float q = 0.f;
  #pragma unroll
  for (int i = 0; i < 4; ++i) { const float d = vals[i] - mean; q += d * d; }
  const float var = block_sum_256(q, sbuf) * (1.0f / DMODEL);
  const float inv = rsqrtf(var + 1e-5f);
  #pragma unroll
  for (int i = 0; i < 4; ++i) {
    const int d = tid + 256 * i;
    const float o = (vals[i] - mean) * inv * g1[d] + be1[d];
    latent[(size_t)blk * DMODEL + d] = o;
    if (n == 0) lat0_out[(size_t)b * DMODEL + d] = o;
  }
}

// ---------------------------------------------------------------------------
// 3) xd = latent @ W_dec + b_dec  via V_WMMA_F32_16X16X4_F32
//    256 thr (8 waves), 128x64 tile, K-chunk 32, double-buffered LDS fed by
//    async global->LDS (ASYNCcnt). B staged in k-paired layout so every
//    fragment is one contiguous 8B ds_load (no register shuffles).
// ---------------------------------------------------------------------------
__global__ __launch_bounds__(256) void gemm_wmma_f32(
    const float* __restrict__ A, const float* __restrict__ W,
    const float* __restrict__ bias, float* __restrict__ C, int R) {
  constexpr int TM = 128, TN = 64, KC = 32, LDA = 36;
  constexpr int LDBP = 160;                 // words per k-pair row (+32 banks)
  __shared__ float sA[2][TM * LDA];         // 2 x 18.0 KB (padded, b64 frags)
  __shared__ float sB[2][(KC / 2) * LDBP];  // 2 x 10.0 KB k-paired layout:
                                            //  (k,n) -> (k>>1)*160 + 2n + (k&1)
  const int tid  = threadIdx.x;
  const int lane = tid & 31;
  const int wave = tid >> 5;
  const int row0 = blockIdx.y * TM;
  const int col0 = blockIdx.x * TN;

  v8f acc[4] = {};

  const int ar = tid >> 3;                  // A: row 0..31 (+32*i)
  const int ak = (tid & 7) << 2;            //     k 0,4,..,28 (16B)

  const int m    = lane & 15;               // fragment row/col within tile
  const int koff = (lane >> 4) << 1;        // lanes 16..31 hold K+2,K+3

  // issue one K-chunk of async copies into buffer p
  auto issue_chunk = [&](int p, int k0) {
    #pragma unroll
    for (int i = 0; i < 4; ++i) {
      const int r  = ar + 32 * i;
      int gr = row0 + r;
      if (gr >= R) gr = R - 1;              // clamp: feeds discarded rows only
      async_copy_b128(lds_off(&sA[p][r * LDA + ak]),
                      A + (size_t)gr * DMODEL + k0 + ak);
    }
    // B: 32x64 chunk, one dword per copy, scattered into k-paired layout
    #pragma unroll
    for (int i = 0; i < 8; ++i) {
      const int e  = i * 256 + tid;         // 0..2047
      const int kk = e >> 6;                // 0..31
      const int nn = e & 63;                // 0..63 (lane-contiguous in global)
      async_copy_b32(lds_off(&sB[p][(kk >> 1) * LDBP + 2 * nn + (kk & 1)]),
                     W + (size_t)(k0 + kk) * DMODEL + col0 + nn);
    }
  };

  issue_chunk(0, 0);

  for (int k0 = 0, it = 0; k0 < DMODEL; k0 += KC, ++it) {
    const int p = it & 1;
    wait_asynccnt0();                       // our chunk-it copies landed
    __syncthreads();                        // everyone's copies landed; prior
                                            // buffer fully consumed by all
    if (k0 + KC < DMODEL) issue_chunk(p ^ 1, k0 + KC);

    const float* aRow  = &sA[p][(wave * 16 + m) * LDA + koff];
    const float* bBase = &sB[p][((koff >> 1) * LDBP) + 2 * m];
    #pragma unroll
    for (int kk = 0; kk < KC; kk += 4) {
      const v2f a = *(const v2f*)(aRow + kk);            // ds_load_2addr_b64
      const float* bRow = bBase + (kk >> 1) * LDBP;
      #pragma unroll
      for (int j = 0; j < 4; ++j) {
        const v2f b = *(const v2f*)(bRow + 32 * j);      // ds_load_2addr_b64
        acc[j] = __builtin_amdgcn_wmma_f32_16x16x4_f32(
            false, a, false, b, (short)0, acc[j], false, false);
      }
    }
  }

  const int half = lane >> 4;
  #pragma unroll
  for (int j = 0; j < 4; ++j) {
    const int col = col0 + j * 16 + m;
    const float bv = bias[col];
    #pragma unroll
    for (int v = 0; v < 8; ++v) {
      const int row = row0 + wave * 16 + half * 8 + v;
      if (row < R) C[(size_t)row * DMODEL + col] = acc[j][v] + bv;
    }
  }
}

// ---------------------------------------------------------------------------
// 4) p_m = LN(mask_token; gamma2,beta2) . W_pred + b_pred   (one block)
// ---------------------------------------------------------------------------
__global__ __launch_bounds__(256) void pm_kernel(
    const float* __restrict__ mtok, const float* __restrict__ g2,
    const float* __restrict__ be2, const float* __restrict__ Wp,
    const float* __restrict__ bp, float* __restrict__ pm) {
  __shared__ float sbuf[256];
  const int tid = threadIdx.x;
  float vals[4];
  #pragma unroll
  for (int i = 0; i < 4; ++i) vals[i] = mtok[tid + 256 * i];
  float s = vals[0] + vals[1] + vals[2] + vals[3];
  const float mean = block_sum_256(s, sbuf) * (1.0f / DMODEL);
  float q = 0.f;
  #pragma unroll
  for (int i = 0; i < 4; ++i) { const float d = vals[i] - mean; q += d * d; }
  const float var = block_sum_256(q, sbuf) * (1.0f / DMODEL);
  const float inv = rsqrtf(var + 1e-5f);
  float local = 0.f;
  #pragma unroll
  for (int i = 0; i < 4; ++i) {
    const int d = tid + 256 * i;
    local += ((vals[i] - mean) * inv * g2[d] + be2[d]) * Wp[d];
  }
  const float tot = block_sum_256(local, sbuf);
  if (tid == 0) pm[0] = tot + bp[0];
}

// ---------------------------------------------------------------------------
// 5) kept positions: pred[b, shuffle[b,n]] = LN2(xd[b,1+n]) . W_pred + b_pred
// ---------------------------------------------------------------------------
__global__ __launch_bounds__(256) void pred_kept_kernel(
    const float* __restrict__ xd, const float* __restrict__ g2,
    const float* __restrict__ be2, const float* __restrict__ Wp,
    const float* __restrict__ bp, const int* __restrict__ shuffle,
    float* __restrict__ pred_out) {
  __shared__ float sbuf[256];
  const int blk = blockIdx.x;               // 0..BATCH*NKEEP-1
  const int b = blk / NKEEP;
  const int n = blk % NKEEP;
  const int tid = threadIdx.x;
  const float* row = xd + (size_t)(b * (NKEEP + 1) + 1 + n) * DMODEL;

  float vals[4];
  #pragma unroll
  for (int i = 0; i < 4; ++i) vals[i] = row[tid + 256 * i];
  float s = vals[0] + vals[1] + vals[2] + vals[3];
  const float mean = block_sum_256(s, sbuf) * (1.0f / DMODEL);
  float q = 0.f;
  #pragma unroll
  for (int i = 0; i < 4; ++i) { const float d = vals[i] - mean; q += d * d; }
  const float var = block_sum_256(q, sbuf) * (1.0f / DMODEL);
  const float inv = rsqrtf(var + 1e-5f);
  float local = 0.f;
  #pragma unroll
  for (int i = 0; i < 4; ++i) {
    const int d = tid + 256 * i;
    local += ((vals[i] - mean) * inv * g2[d] + be2[d]) * Wp[d];
  }
  const float tot = block_sum_256(local, sbuf);
  if (tid == 0) {
    const int l = shuffle[b * SEQ + n];
    pred_out[b * SEQ + l] = tot + bp[0];
  }
}

// ---------------------------------------------------------------------------
// 6) masked positions: pred[b, shuffle[b,n>=500]] = p_m
// ---------------------------------------------------------------------------
__global__ void mask_fill_kernel(const int* __restrict__ shuffle,
                                 const float* __restrict__ pm,
                                 float* __restrict__ pred_out) {
  const int t = blockIdx.x * blockDim.x + threadIdx.x;
  if (t >= BATCH * NMASK) return;
  const int b = t / NMASK;
  const int n = NKEEP + (t % NMASK);
  pred_out[b * SEQ + shuffle[b * SEQ + n]] = pm[0];
}

// ---------------------------------------------------------------------------
// 7) loss = sum(mask * (pred - nan_to_num(expr))^2) / sum(mask)
// ---------------------------------------------------------------------------
__global__ __launch_bounds__(256) void loss_partial_kernel(
    const float* __restrict__ pred, const float* __restrict__ mask,
    const float* __restrict__ expr, float* __restrict__ partials) {
  __shared__ float sbuf[256];
  float s = 0.f, sm = 0.f;
  for (int t = blockIdx.x * blockDim.x + threadIdx.x; t < BATCH * SEQ;
       t += gridDim.x * blockDim.x) {
    const float m = mask[t];
    float e = expr[t];
    e = isnan(e) ? 0.f : e;
    const float d = pred[t] - e;
    s += m * d * d;
    sm += m;
  }
  const float ts = block_sum_256(s, sbuf);
  const float tm = block_sum_256(sm, sbuf);
  if (threadIdx.x == 0) {
    partials[2 * blockIdx.x + 0] = ts;
    partials[2 * blockIdx.x + 1] = tm;
  }
}

__global__ __launch_bounds__(256) void loss_final_kernel(
    const float* __restrict__ partials, float* __restrict__ loss_out) {
  __shared__ float sbuf[256];
  const int tid = threadIdx.x;
  float s = (tid < LOSS_NB) ? partials[2 * tid + 0] : 0.f;
  float m = (tid < LOSS_NB) ? partials[2 * tid + 1] : 0.f;
  const float ts = block_sum_256(s, sbuf);
  const float tm = block_sum_256(m, sbuf);
  if (tid == 0) loss_out[0] = ts / tm;
}

// ---------------------------------------------------------------------------
extern "C" void kernel_launch(void* const* d_in, const int* in_sizes, int n_in,
                              void* d_out, int out_size, void* d_ws,
                              size_t ws_size, hipStream_t stream) {
  (void)in_sizes; (void)n_in; (void)out_size; (void)ws_size;
  const float* expr  = (const float*)d_in[0];
  const int*   idx   = (const int*)d_in[1];
  const float* noise = (const float*)d_in[2];
  const float* pos   = (const float*)d_in[3];
  const float* cls   = (const float*)d_in[4];
  const float* w_enc = (const float*)d_in[5];
  const float* b_enc = (const float*)d_in[6];
  const float* g1    = (const float*)d_in[7];
  const float* be1   = (const float*)d_in[8];
  const float* Wdec  = (const float*)d_in[9];
  const float* bdec  = (const float*)d_in[10];
  const float* mtok  = (const float*)d_in[11];
  const float* g2    = (const float*)d_in[12];
  const float* be2   = (const float*)d_in[13];
  const float* Wp    = (const float*)d_in[14];
  const float* bp    = (const float*)d_in[15];

  // output layout: [loss(1) | pred(64000) | mask(64000) | latent0(32768)]
  float* out      = (float*)d_out;
  float* loss_out = out;
  float* pred_out = out + 1;
  float* mask_out = out + 1 + BATCH * SEQ;
  float* lat0_out = out + 1 + 2 * BATCH * SEQ;

  // workspace layout
  float* latent   = (float*)d_ws;
  float* xd       = latent + (size_t)NROWS * DMODEL;
  float* pm       = xd + (size_t)NROWS * DMODEL;
  float* partials = pm + 4;
  int*   shuffle  = (int*)(partials + 2 * LOSS_NB);
  int*   restore  = shuffle + BATCH * SEQ;

  rank_kernel<<<BATCH, 256, 0, stream>>>(noise, shuffle, restore, mask_out);

  build_latent_kernel<<<NROWS, 256, 0, stream>>>(
      expr, idx, pos, cls, w_enc, b_enc, g1, be1, shuffle, latent, lat0_out);

  dim3 ggrid(DMODEL / 64, (NROWS + 127) / 128);
  gemm_wmma_f32<<<ggrid, 256, 0, stream>>>(latent, Wdec, bdec, xd, NROWS);

  pm_kernel<<<1, 256, 0, stream>>>(mtok, g2, be2, Wp, bp, pm);

  pred_kept_kernel<<<BATCH * NKEEP, 256, 0, stream>>>(
      xd, g2, be2, Wp, bp, shuffle, pred_out);

  mask_fill_kernel<<<(BATCH * NMASK + 255) / 256, 256, 0, stream>>>(
      shuffle, pm, pred_out);

  loss_partial_kernel<<<LOSS_NB, 256, 0, stream>>>(
      pred_out, mask_out, expr, partials);

  loss_final_kernel<<<1, 256, 0, stream>>>(partials, loss_out);
}